// Patcher_15633680957618
// MI455X (gfx1250) — compile-verified
//
#include <hip/hip_runtime.h>
#include <hip/hip_bf16.h>
#include <math.h>

typedef __attribute__((ext_vector_type(2))) float v2f;
typedef __attribute__((ext_vector_type(8))) float v8f;
typedef __attribute__((ext_vector_type(4))) unsigned int u32x4;
typedef __attribute__((ext_vector_type(4))) int i32x4;
typedef __attribute__((ext_vector_type(8))) int i32x8;

#define NEMBD 768
#define BEMB  384
#define KSZ   8
#define TLEN  1024
#define NB    8
#define PMAX  10
#define T2    1026
#define NROWS (NB * TLEN)            // 8192 GEMM rows
#define K1    (PMAX * NEMBD)         // 7680
#define KCONV (BEMB * KSZ)           // 3072
#define KC    64

#define USE_TDM 1

#if USE_TDM
// One-instruction 2D tile DMA: global (stride elems of 4B) -> contiguous LDS.
// D# layout per CDNA5 ISA. Issue from ONE wave.
__device__ __forceinline__ void tdm_load_2d_f32(const void* gaddr, unsigned int lds_addr,
                                                int tile_w, int tile_h,
                                                int tensor_w, int tensor_h, int stride) {
  unsigned long long ga = (unsigned long long)(uintptr_t)gaddr;
  u32x4 g0;
  g0[0] = 1u;                                    // count=1 (valid), no gather/restore
  g0[1] = lds_addr;                              // LDS byte address
  g0[2] = (unsigned int)ga;                      // global_addr[31:0]
  g0[3] = (unsigned int)(ga >> 32) | 0x80000000u; // global_addr[56:32] | type=2
  i32x8 g1;
  g1[0] = 0x00020000;                            // data_size=2 (4B); mask/flags=0
  g1[1] = (tensor_w & 0xFFFF) << 16;             // tensor_dim0[15:0] @bit48
  g1[2] = ((tensor_w >> 16) & 0xFFFF) | ((tensor_h & 0xFFFF) << 16); // dim0 hi | dim1 lo
  g1[3] = ((tensor_h >> 16) & 0xFFFF) | (tile_w << 16);              // dim1 hi | tile_dim0
  g1[4] = tile_h;                                // tile_dim1; tile_dim2=0
  g1[5] = stride;                                // tensor_dim0_stride[31:0] (elems)
  g1[6] = 0;
  g1[7] = 0;
  i32x4 gz = {0, 0, 0, 0};
#if defined(__clang_major__) && __clang_major__ >= 23
  i32x8 gz8 = {0, 0, 0, 0, 0, 0, 0, 0};
  __builtin_amdgcn_tensor_load_to_lds(g0, g1, gz, gz, gz8, 0);
#else
  __builtin_amdgcn_tensor_load_to_lds(g0, g1, gz, gz, 0);
#endif
}
#endif

// ---------------------------------------------------------------- embedding
__global__ __launch_bounds__(256) void embed_kernel(const int* __restrict__ idx,
                                                    const float* __restrict__ wte,
                                                    float* __restrict__ emb) {
  int row = blockIdx.x;                       // (b*1024 + t)
  int id = idx[row];
  const float* src = wte + (size_t)id * NEMBD;
  float* dst = emb + (size_t)row * NEMBD;
  for (int e = threadIdx.x; e < NEMBD; e += 256) dst[e] = src[e];
}

// --------------------------------------- conv1d as WMMA GEMM (pred matrix)
// pred[b,o,u] = sum_{i,k} conv_w[o,i,k] * x[b,i,u-7+k],  x[b,i,t]=emb[b,t,i]
__global__ __launch_bounds__(256) void conv_pred_kernel(const float* __restrict__ emb,
                                                        const float* __restrict__ conv_w,
                                                        float* __restrict__ pred) {
  const int o0 = blockIdx.x * 32;
  const int u0 = blockIdx.y * 128;
  const int b  = blockIdx.z;
  const int tid = threadIdx.x;
  const int lane = tid & 31;
  const int wave = tid >> 5;

  __shared__ __align__(16) float As[32][66];
  __shared__ __align__(16) float xs[8][136];  // channels i0..i0+7, tau = u0-7 .. u0+127

  v8f c0 = {}, c1 = {};
  const int arow = lane & 15;
  const int off  = (lane >> 4) << 1;
  const int wcol = (wave << 4) + (lane & 15);

  for (int k0 = 0; k0 < KCONV; k0 += KC) {
    { // stage A: contiguous rows of conv_w
      int r = tid >> 4;
      int cbase = (tid & 15) << 2;
#pragma unroll
      for (int rr = 0; rr < 32; rr += 16) {
        float4 v = *(const float4*)&conv_w[(size_t)(o0 + r + rr) * KCONV + k0 + cbase];
        As[r + rr][cbase + 0] = v.x;
        As[r + rr][cbase + 1] = v.y;
        As[r + rr][cbase + 2] = v.z;
        As[r + rr][cbase + 3] = v.w;
        if (k0 + KC < KCONV)
          __builtin_prefetch(&conv_w[(size_t)(o0 + r + rr) * KCONV + k0 + KC + cbase], 0, 1);
      }
    }
    { // stage x window for the 8 channels of this K-chunk
      int i0 = k0 >> 3;                       // KC=64 -> 8 channels per chunk
      for (int q = tid; q < 8 * 135; q += 256) {
        int ic = q & 7, w = q >> 3;
        int tau = u0 - 7 + w;
        float v = 0.0f;
        if (tau >= 0 && tau < TLEN) v = emb[((size_t)(b * TLEN + tau)) * NEMBD + i0 + ic];
        xs[ic][w] = v;
      }
    }
    __syncthreads();
#pragma unroll
    for (int kk = 0; kk < KC; kk += 4) {
      int krow = kk + off;                    // even -> (krow&7) <= 6, +1 stays in channel
      int il = krow >> 3;
      int kt = krow & 7;
      v2f bv;
      bv.x = xs[il][kt + wcol];
      bv.y = xs[il][kt + 1 + wcol];
      v2f a0 = *(const v2f*)&As[arow][krow];
      v2f a1 = *(const v2f*)&As[arow + 16][krow];
      c0 = __builtin_amdgcn_wmma_f32_16x16x4_f32(false, a0, false, bv, (short)0, c0, false, false);
      c1 = __builtin_amdgcn_wmma_f32_16x16x4_f32(false, a1, false, bv, (short)0, c1, false, false);
    }
    __syncthreads();
  }

  const int ug = u0 + wcol;
  const int mbase = (lane >> 4) << 3;
#pragma unroll
  for (int v = 0; v < 8; ++v) {
    int o = o0 + mbase + v;
    pred[((size_t)(b * BEMB + o)) * TLEN + ug] = c0[v];
    pred[((size_t)(b * BEMB + o + 16)) * TLEN + ug] = c1[v];
  }
}

// ------------------------------------ losses: deterministic tree reduction
__global__ __launch_bounds__(512) void loss_reduce_kernel(const float* __restrict__ emb,
                                                          const float* __restrict__ pred,
                                                          float* __restrict__ losses) {
  const int t = blockIdx.x;                   // 0..1022
  const int b = blockIdx.y;
  __shared__ float red[512];
  int c = threadIdx.x;
  float v = 0.0f;
  if (c < BEMB) {
    float x = emb[((size_t)(b * TLEN + t)) * NEMBD + c];
    float p = pred[((size_t)(b * BEMB + c)) * TLEN + t + 1];
    float d = x - p;
    v = d * d;
  }
  red[c] = v;
  __syncthreads();
  for (int s = 256; s > 0; s >>= 1) {
    if (c < s) red[c] += red[c + s];
    __syncthreads();
  }
  if (c == 0) losses[b * (TLEN - 1) + t] = red[0] * (1.0f / BEMB);
}

// ------------------------------------------------------------- init buffers
__global__ void fill_kernel(int* __restrict__ tok, float* __restrict__ tgt) {
  int i = blockIdx.x * blockDim.x + threadIdx.x;
  if (i < NB * T2 * PMAX) tok[i] = -1;
  if (i < NB * TLEN * PMAX) tgt[i] = -1.0f;
}

// ------------------------------------------------------- sequential scan
__global__ void scan_kernel(const float* __restrict__ losses, const int* __restrict__ idx,
                            const float* __restrict__ threshold,
                            int* __restrict__ tok, float* __restrict__ tgt) {
  int b = threadIdx.x;
  if (b >= NB) return;
  const float thr = threshold[0];
  float acc = 0.0f;
  int dep = 0, ln = 0;
  for (int t = 0; t < TLEN - 1; ++t) {
    acc += losses[b * (TLEN - 1) + t];
    bool mask = (acc > thr) || (ln >= PMAX - 1);
    dep += mask ? 1 : 0;
    ln = mask ? 0 : (ln + 1);
    if (mask) acc = 0.0f;
    tok[(b * T2 + dep) * PMAX + ln] = t;      // unique slot per t
    if (dep >= 1 && dep <= TLEN) {            // output keeps patches 1..1024
      size_t base = ((size_t)(b * TLEN) + (dep - 1)) * PMAX;
      if (ln + 1 < PMAX) tgt[base + ln + 1] = 50256.0f;   // VOCAB-1 marker
      tgt[base + ln] = (float)idx[b * TLEN + t];          // token overwrites marker
    }
  }
}

// ---------------------------------------------- GEMM1: gathered A @ w1, GELU
// M=8192 (b,p), K=7680, N=768; block 32x128; B tile staged by TDM (1 DMA/stage).
__global__ __launch_bounds__(256) void gemm1_kernel(const float* __restrict__ emb,
                                                    const int* __restrict__ tok,
                                                    const float* __restrict__ wpe,
                                                    const float* __restrict__ w1,
                                                    const float* __restrict__ b1,
                                                    float* __restrict__ h) {
  const int m0 = blockIdx.x * 32;
  const int n0 = blockIdx.y * 128;
  const int tid = threadIdx.x;
  const int lane = tid & 31;
  const int wave = tid >> 5;
  const int b = m0 >> 10;
  const int p0 = m0 & (TLEN - 1);

  __shared__ __align__(16) float As[32][66];
  __shared__ __align__(16) float Bs[KC][128];

  v8f c0 = {}, c1 = {};
  const int arow = lane & 15;
  const int off = (lane >> 4) << 1;
  const int wcol = (wave << 4) + (lane & 15);

  for (int k0 = 0; k0 < K1; k0 += KC) {
    { // stage A (gather through tok map, add wpe)
      int r = tid >> 4;
      int cbase = (tid & 15) << 2;
      int kg = k0 + cbase;
      int s = kg / NEMBD;
      int e0 = kg % NEMBD;
      float4 pv = *(const float4*)&wpe[s * NEMBD + e0];
#pragma unroll
      for (int rr = 0; rr < 32; rr += 16) {
        int p = p0 + r + rr;
        int t = tok[(b * T2 + p) * PMAX + s];
        float4 ev = make_float4(0.f, 0.f, 0.f, 0.f);
        if (t >= 0) ev = *(const float4*)&emb[((size_t)(b * TLEN + t)) * NEMBD + e0];
        As[r + rr][cbase + 0] = ev.x + pv.x;
        As[r + rr][cbase + 1] = ev.y + pv.y;
        As[r + rr][cbase + 2] = ev.z + pv.z;
        As[r + rr][cbase + 3] = ev.w + pv.w;
      }
    }
#if USE_TDM
    if (tid < 32) {   // wave 0 issues one tensor DMA for the whole 64x128 B tile
      tdm_load_2d_f32(&w1[(size_t)k0 * NEMBD + n0],
                      (unsigned int)(uintptr_t)&Bs[0][0],
                      128, KC, NEMBD, KC, NEMBD);
      __builtin_amdgcn_s_wait_tensorcnt(0);
    }
#else
    {
      int kr = tid >> 2;
      int cs = (tid & 3) << 5;
      const float* src = &w1[(size_t)(k0 + kr) * NEMBD + n0 + cs];
#pragma unroll
      for (int j = 0; j < 32; j += 4) {
        float4 v = *(const float4*)&src[j];
        Bs[kr][cs + j + 0] = v.x;
        Bs[kr][cs + j + 1] = v.y;
        Bs[kr][cs + j + 2] = v.z;
        Bs[kr][cs + j + 3] = v.w;
      }
    }
#endif
    __syncthreads();
#pragma unroll
    for (int kk = 0; kk < KC; kk += 4) {
      v2f bv;
      bv.x = Bs[kk + off][wcol];
      bv.y = Bs[kk + off + 1][wcol];
      v2f a0 = *(const v2f*)&As[arow][kk + off];
      v2f a1 = *(const v2f*)&As[arow + 16][kk + off];
      c0 = __builtin_amdgcn_wmma_f32_16x16x4_f32(false, a0, false, bv, (short)0, c0, false, false);
      c1 = __builtin_amdgcn_wmma_f32_16x16x4_f32(false, a1, false, bv, (short)0, c1, false, false);
    }
    __syncthreads();
  }

  const int colg = n0 + wcol;
  const float bias = b1[colg];
  const int mbase = (lane >> 4) << 3;
#pragma unroll
  for (int v = 0; v < 8; ++v) {
    float x0 = c0[v] + bias;
    float x1 = c1[v] + bias;
    float g0 = 0.5f * x0 * (1.0f + erff(x0 * 0.70710678118654752f));
    float g1 = 0.5f * x1 * (1.0f + erff(x1 * 0.70710678118654752f));
    h[(size_t)(m0 + mbase + v) * NEMBD + colg] = g0;
    h[(size_t)(m0 + 16 + mbase + v) * NEMBD + colg] = g1;
  }
}

// ---------------------------------------------------- GEMM2: h @ w2 + b2
__global__ __launch_bounds__(256) void gemm2_kernel(const float* __restrict__ h,
                                                    const float* __restrict__ w2,
                                                    const float* __restrict__ b2,
                                                    float* __restrict__ outp) {
  const int m0 = blockIdx.x * 32;
  const int n0 = blockIdx.y * 128;
  const int tid = threadIdx.x;
  const int lane = tid & 31;
  const int wave = tid >> 5;

  __shared__ __align__(16) float As[32][66];
  __shared__ __align__(16) float Bs[KC][128];

  v8f c0 = {}, c1 = {};
  const int arow = lane & 15;
  const int off = (lane >> 4) << 1;
  const int wcol = (wave << 4) + (lane & 15);

  for (int k0 = 0; k0 < NEMBD; k0 += KC) {
    {
      int r = tid >> 4;
      int cbase = (tid & 15) << 2;
#pragma unroll
      for (int rr = 0; rr < 32; rr += 16) {
        float4 ev = *(const float4*)&h[(size_t)(m0 + r + rr) * NEMBD + k0 + cbase];
        As[r + rr][cbase + 0] = ev.x;
        As[r + rr][cbase + 1] = ev.y;
        As[r + rr][cbase + 2] = ev.z;
        As[r + rr][cbase + 3] = ev.w;
      }
    }
#if USE_TDM
    if (tid < 32) {
      tdm_load_2d_f32(&w2[(size_t)k0 * NEMBD + n0],
                      (unsigned int)(uintptr_t)&Bs[0][0],
                      128, KC, NEMBD, KC, NEMBD);
      __builtin_amdgcn_s_wait_tensorcnt(0);
    }
#else
    {
      int kr = tid >> 2;
      int cs = (tid & 3) << 5;
      const float* src = &w2[(size_t)(k0 + kr) * NEMBD + n0 + cs];
#pragma unroll
      for (int j = 0; j < 32; j += 4) {
        float4 v = *(const float4*)&src[j];
        Bs[kr][cs + j + 0] = v.x;
        Bs[kr][cs + j + 1] = v.y;
        Bs[kr][cs + j + 2] = v.z;
        Bs[kr][cs + j + 3] = v.w;
      }
    }
#endif
    __syncthreads();
#pragma unroll
    for (int kk = 0; kk < KC; kk += 4) {
      v2f bv;
      bv.x = Bs[kk + off][wcol];
      bv.y = Bs[kk + off + 1][wcol];
      v2f a0 = *(const v2f*)&As[arow][kk + off];
      v2f a1 = *(const v2f*)&As[arow + 16][kk + off];
      c0 = __builtin_amdgcn_wmma_f32_16x16x4_f32(false, a0, false, bv, (short)0, c0, false, false);
      c1 = __builtin_amdgcn_wmma_f32_16x16x4_f32(false, a1, false, bv, (short)0, c1, false, false);
    }
    __syncthreads();
  }

  const int colg = n0 + wcol;
  const float bias = b2[colg];
  const int mbase = (lane >> 4) << 3;
#pragma unroll
  for (int v = 0; v < 8; ++v) {
    outp[(size_t)(m0 + mbase + v) * NEMBD + colg] = c0[v] + bias;
    outp[(size_t)(m0 + 16 + mbase + v) * NEMBD + colg] = c1[v] + bias;
  }
}

// --------------------------------------------------------------- launcher
extern "C" void kernel_launch(void* const* d_in, const int* in_sizes, int n_in,
                              void* d_out, int out_size, void* d_ws, size_t ws_size,
                              hipStream_t stream) {
  (void)in_sizes; (void)n_in; (void)out_size; (void)ws_size;
  const int*   idx    = (const int*)d_in[0];
  const float* wte    = (const float*)d_in[1];
  const float* wpe    = (const float*)d_in[2];
  const float* conv_w = (const float*)d_in[3];
  const float* thr    = (const float*)d_in[4];
  const float* w1     = (const float*)d_in[5];
  const float* b1     = (const float*)d_in[6];
  const float* w2     = (const float*)d_in[7];
  const float* b2     = (const float*)d_in[8];

  char* ws = (char*)d_ws;
  float* emb  = (float*)ws;                                    // 25165824 B
  float* h    = (float*)(ws + (size_t)NROWS * NEMBD * 4);      // 25165824 B
  int*   tok  = (int*)(ws + 2 * (size_t)NROWS * NEMBD * 4);    // 328320 B
  float* pred = (float*)(ws + 2 * (size_t)NROWS * NEMBD * 4 + 328320); // 12582912 B

  float* outp   = (float*)d_out;                               // patch_embeds: 8*1024*768
  float* tgt    = outp + (size_t)NROWS * NEMBD;                // patch_targets: 8*1024*10
  float* losses = tgt + (size_t)NB * TLEN * PMAX;              // losses: 8*1023

  embed_kernel<<<NROWS, 256, 0, stream>>>(idx, wte, emb);
  conv_pred_kernel<<<dim3(BEMB / 32, TLEN / 128, NB), 256, 0, stream>>>(emb, conv_w, pred);
  loss_reduce_kernel<<<dim3(TLEN - 1, NB), 512, 0, stream>>>(emb, pred, losses);
  fill_kernel<<<(NB * T2 * PMAX + 255) / 256, 256, 0, stream>>>(tok, tgt);
  scan_kernel<<<1, NB, 0, stream>>>(losses, idx, thr, tok, tgt);
  gemm1_kernel<<<dim3(NROWS / 32, NEMBD / 128), 256, 0, stream>>>(emb, tok, wpe, w1, b1, h);
  gemm2_kernel<<<dim3(NROWS / 32, NEMBD / 128), 256, 0, stream>>>(h, w2, b2, outp);
}